// Sample_GAT_62053687493145
// MI455X (gfx1250) — compile-verified
//
#include <hip/hip_runtime.h>
#include <math.h>

#define NEG_SLOPE 0.2f
#define EPS_DEN   1e-16f

typedef float v2f __attribute__((ext_vector_type(2)));
typedef float v8f __attribute__((ext_vector_type(8)));

// ---------------------------------------------------------------- fill
__global__ void fill_kernel(float* __restrict__ p, float v, long n) {
  long i = (long)blockIdx.x * blockDim.x + threadIdx.x;
  if (i < n) p[i] = v;
}

// ---------------------------------------------------------------- GEMM (WMMA fp32)
// C[M x Nout] = A[M x K] * W[K x Nout].  M % 16 == 0, K % 4 == 0, Nout % 16 == 0.
// grid.x = M/16; blockDim.x = 32 * (Nout/16); dynamic LDS = 16*(K+4) floats.
__global__ void __launch_bounds__(256)
gemm_wmma_f32(const float* __restrict__ A, const float* __restrict__ W,
              float* __restrict__ C, int K, int Nout) {
  extern __shared__ float ldsA[];          // 16 rows, stride K+4 (bank-conflict pad)
  const int KP = K + 4;
  const long rowBase = (long)blockIdx.x * 16;

  // cooperative, coalesced load of the 16 x K A-tile into LDS
  for (int idx = threadIdx.x; idx < 16 * K; idx += blockDim.x) {
    int r = idx / K, c = idx - r * K;
    ldsA[r * KP + c] = A[(rowBase + r) * K + c];
  }
  __syncthreads();

  const int wave = threadIdx.x >> 5;
  const int lane = threadIdx.x & 31;
  const int half = lane >> 4;              // 0: lanes 0-15, 1: lanes 16-31
  const int l16  = lane & 15;
  const int colBase = wave * 16;

  v8f acc = {};
  for (int k0 = 0; k0 < K; k0 += 4) {
    const int ka = k0 + 2 * half;          // ISA 16x4 fp32 A layout: K={0,1}|{2,3}
    v2f a, b;
    a.x = ldsA[l16 * KP + ka];
    a.y = ldsA[l16 * KP + ka + 1];
    b.x = W[(long)ka * Nout + colBase + l16];
    b.y = W[(long)(ka + 1) * Nout + colBase + l16];
    acc = __builtin_amdgcn_wmma_f32_16x16x4_f32(false, a, false, b,
                                                (short)0, acc, false, false);
  }
  // C/D layout: lanes 0-15 -> M=v, lanes 16-31 -> M=v+8; N = lane&15
  #pragma unroll
  for (int v = 0; v < 8; ++v) {
    C[(rowBase + v + 8 * half) * (long)Nout + colBase + l16] = acc[v];
  }
}

// ---------------------------------------------------------------- per-node alpha dots
__global__ void alpha_kernel(const float* __restrict__ h,
                             const float* __restrict__ av_src,
                             const float* __restrict__ av_dst,
                             float* __restrict__ asrc, float* __restrict__ adst,
                             int N, int F) {
  int node = blockIdx.x * (blockDim.x >> 5) + (threadIdx.x >> 5);
  int lane = threadIdx.x & 31;
  if (node >= N) return;
  float s1 = 0.f, s2 = 0.f;
  for (int f = lane; f < F; f += 32) {
    float hv = h[(long)node * F + f];
    s1 += hv * av_src[f];
    s2 += hv * av_dst[f];
  }
  #pragma unroll
  for (int off = 16; off > 0; off >>= 1) {
    s1 += __shfl_down(s1, off, 32);
    s2 += __shfl_down(s2, off, 32);
  }
  if (lane == 0) { asrc[node] = s1; adst[node] = s2; }
}

// ---------------------------------------------------------------- float atomic max
__device__ inline void atomicMaxF(float* addr, float val) {
  if (val >= 0.f) atomicMax((int*)addr, __float_as_int(val));
  else            atomicMin((unsigned int*)addr, __float_as_uint(val));
}

__device__ inline void edge_endpoints(const long long* __restrict__ ei,
                                      long E, long i, int& s, int& d) {
  if (i < E) { s = (int)ei[i]; d = (int)ei[E + i]; }
  else       { s = d = (int)(i - E); }           // self loop
}

// ---------------------------------------------------------------- pass 1: logits + segment max
__global__ void edge_logit_max(const long long* __restrict__ ei, long E, int N,
                               const float* __restrict__ asrc,
                               const float* __restrict__ adst,
                               float* __restrict__ e, float* __restrict__ emax) {
  long i = (long)blockIdx.x * blockDim.x + threadIdx.x;
  if (i >= E + N) return;
  int s, d; edge_endpoints(ei, E, i, s, d);
  float x = asrc[s] + adst[d];
  x = (x > 0.f) ? x : NEG_SLOPE * x;             // leaky relu
  e[i] = x;
  atomicMaxF(&emax[d], x);
}

// ---------------------------------------------------------------- pass 2: exp + segment sum
__global__ void edge_exp(const long long* __restrict__ ei, long E, int N,
                         float* __restrict__ e, const float* __restrict__ emax,
                         float* __restrict__ denom) {
  long i = (long)blockIdx.x * blockDim.x + threadIdx.x;
  if (i >= E + N) return;
  int s, d; edge_endpoints(ei, E, i, s, d);
  float ex = __expf(e[i] - emax[d]);
  e[i] = ex;                                     // e now holds e_exp
  atomicAdd(&denom[d], ex);
}

// ---------------------------------------------------------------- pass 3: alpha * h[src] scatter
__global__ void aggregate(const long long* __restrict__ ei, long E, int N,
                          const float* __restrict__ e_exp,
                          const float* __restrict__ denom,
                          const float* __restrict__ h,
                          float* __restrict__ out, int F) {
  long gid = (long)blockIdx.x * blockDim.x + threadIdx.x;
  long total = (E + N) * (long)F;
  if (gid >= total) return;
  long edge = gid / F;
  int  f    = (int)(gid - edge * F);             // consecutive threads -> consecutive f
  int s, d; edge_endpoints(ei, E, edge, s, d);
  float alpha = e_exp[edge] / (denom[d] + EPS_DEN);
  atomicAdd(&out[(long)d * F + f], alpha * h[(long)s * F + f]);
}

// ---------------------------------------------------------------- bias (+relu)
__global__ void bias_act(float* __restrict__ p, const float* __restrict__ bias,
                         long n, int F, int relu) {
  long i = (long)blockIdx.x * blockDim.x + threadIdx.x;
  if (i >= n) return;
  float v = p[i] + bias[i % F];
  if (relu) v = fmaxf(v, 0.f);
  p[i] = v;
}

// ================================================================ driver
static inline int cdiv(long a, long b) { return (int)((a + b - 1) / b); }

extern "C" void kernel_launch(void* const* d_in, const int* in_sizes, int n_in,
                              void* d_out, int out_size, void* d_ws, size_t ws_size,
                              hipStream_t stream) {
  const float*     x   = (const float*)d_in[0];
  const long long* ei  = (const long long*)d_in[1];   // int64 (2,E)
  // d_in[2] edge_weight: ignored by GATConv with edge_dim=None
  const float* W1  = (const float*)d_in[3];
  const float* a1s = (const float*)d_in[4];
  const float* a1d = (const float*)d_in[5];
  const float* b1  = (const float*)d_in[6];
  const float* W2  = (const float*)d_in[7];
  const float* a2s = (const float*)d_in[8];
  const float* a2d = (const float*)d_in[9];
  const float* b2  = (const float*)d_in[10];

  const long E  = in_sizes[1] / 2;        // 800000
  const int  H1 = in_sizes[4];            // 128
  const int  DIN = in_sizes[3] / H1;      // 256
  const int  N  = in_sizes[0] / DIN;      // 50000
  const int  H2 = in_sizes[8];            // 16

  float* ws   = (float*)d_ws;
  float* h1   = ws;  ws += (long)N * H1;
  float* agg1 = ws;  ws += (long)N * H1;  // becomes relu(h) -> layer-2 input
  float* h2   = ws;  ws += (long)N * H2;
  float* asrc = ws;  ws += N;
  float* adst = ws;  ws += N;
  float* emax = ws;  ws += N;
  float* den  = ws;  ws += N;
  float* e    = ws;  ws += (E + N);
  float* out  = (float*)d_out;

  const long EN = E + N;

  // ---------------- layer 1 ----------------
  gemm_wmma_f32<<<N / 16, 32 * (H1 / 16), 16 * (DIN + 4) * sizeof(float), stream>>>(
      x, W1, h1, DIN, H1);
  alpha_kernel<<<cdiv(N, 8), 256, 0, stream>>>(h1, a1s, a1d, asrc, adst, N, H1);
  fill_kernel<<<cdiv(N, 256), 256, 0, stream>>>(emax, -INFINITY, N);
  fill_kernel<<<cdiv(N, 256), 256, 0, stream>>>(den, 0.f, N);
  fill_kernel<<<cdiv((long)N * H1, 256), 256, 0, stream>>>(agg1, 0.f, (long)N * H1);
  edge_logit_max<<<cdiv(EN, 256), 256, 0, stream>>>(ei, E, N, asrc, adst, e, emax);
  edge_exp<<<cdiv(EN, 256), 256, 0, stream>>>(ei, E, N, e, emax, den);
  aggregate<<<cdiv(EN * H1, 256), 256, 0, stream>>>(ei, E, N, e, den, h1, agg1, H1);
  bias_act<<<cdiv((long)N * H1, 256), 256, 0, stream>>>(agg1, b1, (long)N * H1, H1, 1);

  // ---------------- layer 2 ----------------
  gemm_wmma_f32<<<N / 16, 32 * (H2 / 16), 16 * (H1 + 4) * sizeof(float), stream>>>(
      agg1, W2, h2, H1, H2);
  alpha_kernel<<<cdiv(N, 8), 256, 0, stream>>>(h2, a2s, a2d, asrc, adst, N, H2);
  fill_kernel<<<cdiv(N, 256), 256, 0, stream>>>(emax, -INFINITY, N);
  fill_kernel<<<cdiv(N, 256), 256, 0, stream>>>(den, 0.f, N);
  fill_kernel<<<cdiv((long)N * H2, 256), 256, 0, stream>>>(out, 0.f, (long)N * H2);
  edge_logit_max<<<cdiv(EN, 256), 256, 0, stream>>>(ei, E, N, asrc, adst, e, emax);
  edge_exp<<<cdiv(EN, 256), 256, 0, stream>>>(ei, E, N, e, emax, den);
  aggregate<<<cdiv(EN * H2, 256), 256, 0, stream>>>(ei, E, N, e, den, h2, out, H2);
  bias_act<<<cdiv((long)N * H2, 256), 256, 0, stream>>>(out, b2, (long)N * H2, H2, 0);
}